// my_cdist_46583215292845
// MI455X (gfx1250) — compile-verified
//
#include <hip/hip_runtime.h>
#include <cstdint>

// Problem constants (match reference)
#define KN 8192
#define KM 1024
#define KD 64
// Tiling
#define BN 64          // rows per block
#define BM 64          // cols per m-tile
#define MT 4           // m-tiles per block -> 256 cols per block, grid.y = 4 (more resident waves/SIMD)
#define PAD 68         // padded LDS row stride in floats (64 + 4) -> conflict-free, 16B aligned

typedef unsigned int u32;
typedef __attribute__((ext_vector_type(4))) u32 u32x4;
typedef __attribute__((ext_vector_type(8))) int i32x8;
typedef __attribute__((ext_vector_type(4))) int i32x4;

#if __has_builtin(__builtin_amdgcn_tensor_load_to_lds) && __has_builtin(__builtin_amdgcn_s_wait_tensorcnt)
#define USE_TDM 1
#else
#define USE_TDM 0
#endif

__device__ __forceinline__ u32 lds_off(const void* p) {
  // Flat shared-space address: low 32 bits are the workgroup-relative LDS byte address.
  return (u32)(uintptr_t)p;
}

#if USE_TDM
// TDM: load a 64x64 fp32 tile (row stride 64 floats in memory) into LDS with
// 4-dword padding inserted after every 64 dwords -> LDS layout [64][68] floats.
__device__ __forceinline__ void tdm_load_w_tile(const float* gsrc, u32 lds_byte_addr) {
  uint64_t ga = (uint64_t)(uintptr_t)gsrc;
  u32x4 g0;
  g0[0] = 1u;                                             // count=1 (valid), user mode
  g0[1] = lds_byte_addr;                                  // lds_addr
  g0[2] = (u32)(ga & 0xFFFFFFFFull);                      // global_addr[31:0]
  g0[3] = (u32)((ga >> 32) & 0x1FFFFFFull) | (2u << 30);  // global_addr[56:32] | type=2
  i32x8 g1;
  g1[0] = (int)((2u << 16)     // data_size = 4 bytes
              | (1u << 20)     // pad_enable
              | (5u << 22)     // pad_interval: 64 dwords between pads
              | (3u << 25));   // pad_amount: 4 dwords
  g1[1] = (int)(64u << 16);    // tensor_dim0 = 64 (bits 63:48)
  g1[2] = (int)(64u << 16);    // tensor_dim1 = 64 (bits 95:80)
  g1[3] = (int)(64u << 16);    // tile_dim0   = 64 (bits 127:112)
  g1[4] = (int)64;             // tile_dim1   = 64 (bits 143:128), tile_dim2 = 0
  g1[5] = (int)64;             // tensor_dim0_stride = 64 (bits 191:160)
  g1[6] = 0;                   // stride hi / dim1_stride lo = 0
  g1[7] = 0;
  i32x4 z4 = {0, 0, 0, 0};
#if defined(__clang_major__) && (__clang_major__ >= 23)
  i32x8 z8 = {0, 0, 0, 0, 0, 0, 0, 0};
  __builtin_amdgcn_tensor_load_to_lds(g0, g1, z4, z4, z8, 0);
#else
  __builtin_amdgcn_tensor_load_to_lds(g0, g1, z4, z4, 0);
#endif
}
#endif

__device__ __forceinline__ float l1_4(float4 a, float4 b) {
  // Balanced tree: 4x v_sub + 3x v_add with free |src| modifiers (+1 add into acc at call site)
  return (fabsf(a.x - b.x) + fabsf(a.y - b.y)) + (fabsf(a.z - b.z) + fabsf(a.w - b.w));
}

__global__ __launch_bounds__(256)
void cdist_l1_kernel(const float* __restrict__ x, const float* __restrict__ w,
                     float* __restrict__ out) {
  __shared__ float xs[BN][PAD];        // 17,408 B
  __shared__ float ws[2][BM][PAD];     // 34,816 B (double-buffered weight tiles)

  const int tid = threadIdx.x;
  const int r0  = blockIdx.x * BN;
  const int c0  = blockIdx.y * (BM * MT);
  const int tx  = tid & 15;            // m direction
  const int ty  = tid >> 4;            // n direction
  const bool wave0 = (tid < 32);       // wave-uniform predicate

  // Stage x tile: 64 rows x 64 floats = 1024 float4, 4 per thread.
  #pragma unroll
  for (int i = 0; i < 4; ++i) {
    int idx = tid + i * 256;
    int row = idx >> 4;
    int c4  = (idx & 15) << 2;
    float4 v = *(const float4*)(x + (size_t)(r0 + row) * KD + c4);
    *(float4*)(&xs[row][c4]) = v;
  }

#if USE_TDM
  if (wave0) tdm_load_w_tile(w + (size_t)c0 * KD, lds_off(&ws[0][0][0]));
#endif

  for (int t = 0; t < MT; ++t) {
#if USE_TDM
    if (wave0) __builtin_amdgcn_s_wait_tensorcnt(0);   // buf[t&1] ready (wave0 holds TENSORcnt)
    __syncthreads();                                    // publish tile + xs; prior compute done
    if (wave0 && (t + 1 < MT))                          // prefetch next tile into the other buffer
      tdm_load_w_tile(w + (size_t)(c0 + (t + 1) * BM) * KD,
                      lds_off(&ws[(t + 1) & 1][0][0]));
    const float (*wt)[PAD] = (const float (*)[PAD])ws[t & 1];
#else
    // Fallback: cooperative global->LDS staging (single buffer).
    __syncthreads();                                    // previous compute done before overwrite
    #pragma unroll
    for (int i = 0; i < 4; ++i) {
      int idx = tid + i * 256;
      int row = idx >> 4;
      int c4  = (idx & 15) << 2;
      float4 v = *(const float4*)(w + (size_t)(c0 + t * BM + row) * KD + c4);
      *(float4*)(&ws[0][row][c4]) = v;
    }
    __syncthreads();
    const float (*wt)[PAD] = (const float (*)[PAD])ws[0];
#endif

    float acc[4][4] = {};
    for (int d4 = 0; d4 < KD; d4 += 4) {
      float4 xv[4], wv[4];
      #pragma unroll
      for (int ii = 0; ii < 4; ++ii) xv[ii] = *(const float4*)(&xs[ty + ii * 16][d4]);
      #pragma unroll
      for (int jj = 0; jj < 4; ++jj) wv[jj] = *(const float4*)(&wt[tx + jj * 16][d4]);
      #pragma unroll
      for (int ii = 0; ii < 4; ++ii)
        #pragma unroll
        for (int jj = 0; jj < 4; ++jj)
          acc[ii][jj] += l1_4(xv[ii], wv[jj]);   // v_sub_f32 + v_add_f32 with |src| modifiers
    }

    const int mb = c0 + t * BM;
    #pragma unroll
    for (int ii = 0; ii < 4; ++ii)
      #pragma unroll
      for (int jj = 0; jj < 4; ++jj)
        out[(size_t)(r0 + ty + ii * 16) * KM + (mb + tx + jj * 16)] = -acc[ii][jj];
    // No trailing barrier needed: next iteration's wait+barrier precedes buffer reuse.
  }
}

extern "C" void kernel_launch(void* const* d_in, const int* in_sizes, int n_in,
                              void* d_out, int out_size, void* d_ws, size_t ws_size,
                              hipStream_t stream) {
  (void)in_sizes; (void)n_in; (void)d_ws; (void)ws_size; (void)out_size;
  const float* x = (const float*)d_in[0];   // input  [8192, 64]
  const float* w = (const float*)d_in[1];   // weight [1024, 64]
  float* out = (float*)d_out;               // out    [8192, 1024]
  dim3 grid(KN / BN, KM / (BM * MT));       // (128, 4)
  cdist_l1_kernel<<<grid, 256, 0, stream>>>(x, w, out);
}